// Net_87540023427398
// MI455X (gfx1250) — compile-verified
//
#include <hip/hip_runtime.h>
#include <math.h>

// GCN (2-layer, PyG gcn_norm) for MI455X / gfx1250.
// Dense transform x@W1 uses V_WMMA_F32_16X16X4_F32 (exact fp32 math).
// Edge gather/scatter uses f32 global atomics; node arrays (~6.4MB) live in
// the 192MB L2, so the scatter phase is L2-resident, HBM only streams edges.

typedef __attribute__((ext_vector_type(2))) float v2f;
typedef __attribute__((ext_vector_type(8))) float v8f;

#define FIN 37
#define HID 16
#define NCLS 2

__global__ void zero_kernel(float* __restrict__ p, long n) {
  long i = (long)blockIdx.x * blockDim.x + threadIdx.x;
  if (i < n) p[i] = 0.0f;
}

// deg[t] += w[e] for each edge (self-loop +1 added in dinv_kernel)
__global__ void deg_kernel(const int* __restrict__ col, const float* __restrict__ w,
                           float* __restrict__ deg, int E) {
  int e = blockIdx.x * blockDim.x + threadIdx.x;
  if (e < E) atomicAdd(&deg[col[e]], w[e]);
}

// in-place: deg -> D^-1/2 (with self-loop weight 1.0 included)
__global__ void dinv_kernel(float* __restrict__ deg, int n) {
  int i = blockIdx.x * blockDim.x + threadIdx.x;
  if (i < n) {
    float d = deg[i] + 1.0f;
    deg[i] = (d > 0.0f) ? rsqrtf(d) : 0.0f;
  }
}

// h = x @ W1   (N x 37) @ (37 x 16), one wave per 16-row tile via
// V_WMMA_F32_16X16X4_F32; K padded 37 -> 40 with zeros (10 WMMA per tile).
// A layout (16x4 f32): lane l holds A[l%16][2*(l/16)+{0,1}] in {a.x,a.y}.
// B layout (4x16 f32, mirrored): lane l holds B[2*(l/16)+{0,1}][l%16].
// C/D (16x16 f32): vgpr r, lane l -> D[r + 8*(l/16)][l%16].
__global__ void xw1_wmma_kernel(const float* __restrict__ x, const float* __restrict__ W1,
                                float* __restrict__ h, int nTiles) {
  int wave = (int)((blockIdx.x * (unsigned)blockDim.x + threadIdx.x) >> 5);
  if (wave >= nTiles) return;                 // wave-uniform: EXEC stays all-1s
  int lane = threadIdx.x & 31;
  int ln = lane & 15;                         // A row index / B col index
  int kh = lane >> 4;                         // which K half this lane holds
  int m0 = wave * 16;
  v8f c = {};
#pragma unroll
  for (int k0 = 0; k0 < 40; k0 += 4) {
    int ka = k0 + 2 * kh;
    v2f a, b;
    a.x = (ka     < FIN) ? x[(long)(m0 + ln) * FIN + ka]     : 0.0f;
    a.y = (ka + 1 < FIN) ? x[(long)(m0 + ln) * FIN + ka + 1] : 0.0f;
    b.x = (ka     < FIN) ? W1[ka * HID + ln]                 : 0.0f;
    b.y = (ka + 1 < FIN) ? W1[(ka + 1) * HID + ln]           : 0.0f;
    c = __builtin_amdgcn_wmma_f32_16x16x4_f32(false, a, false, b,
                                              (short)0, c, false, false);
  }
#pragma unroll
  for (int r = 0; r < 8; ++r)
    h[(long)(m0 + r + 8 * kh) * HID + ln] = c[r];
}

// scalar fallback for a ragged row tail (not used when N % 16 == 0)
__global__ void xw1_tail_kernel(const float* __restrict__ x, const float* __restrict__ W1,
                                float* __restrict__ h, int i0, int n) {
  int idx = blockIdx.x * blockDim.x + threadIdx.x;
  int i = i0 + (idx >> 4);
  int j = idx & 15;
  if (i >= n) return;
  float acc = 0.0f;
  for (int k = 0; k < FIN; ++k) acc += x[(long)i * FIN + k] * W1[k * HID + j];
  h[(long)i * HID + j] = acc;
}

// agg1[t][:] += dinv[s]*w*dinv[t] * h[s][:]  (edges only; self-loop fused later)
__global__ void agg1_kernel(const int* __restrict__ row, const int* __restrict__ col,
                            const float* __restrict__ w, const float* __restrict__ dinv,
                            const float* __restrict__ h, float* __restrict__ agg, int E) {
  int e = blockIdx.x * blockDim.x + threadIdx.x;
  if (e >= E) return;
  int s = row[e], t = col[e];
  float coef = dinv[s] * w[e] * dinv[t];
  const float* hs = h + (long)s * HID;
  float* at = agg + (long)t * HID;
#pragma unroll
  for (int f = 0; f < HID; ++f) atomicAdd(&at[f], coef * hs[f]);
}

// h1 = relu(agg1 + dinv^2 * h + b1), written in place over h
__global__ void relu_bias_kernel(const float* __restrict__ agg, const float* __restrict__ dinv,
                                 const float* __restrict__ b1, float* __restrict__ h, long n16) {
  long idx = (long)blockIdx.x * blockDim.x + threadIdx.x;
  if (idx >= n16) return;
  long i = idx >> 4;
  int f = (int)(idx & 15);
  float di = dinv[i];
  float v = agg[idx] + di * di * h[idx] + b1[f];
  h[idx] = v > 0.0f ? v : 0.0f;
}

// g = h1 @ W2   (N x 16) @ (16 x 2): too skinny for WMMA, scalar FMA
__global__ void gw2_kernel(const float* __restrict__ h1, const float* __restrict__ W2,
                           float* __restrict__ g, int n) {
  int i = blockIdx.x * blockDim.x + threadIdx.x;
  if (i >= n) return;
  float z0 = 0.0f, z1 = 0.0f;
#pragma unroll
  for (int k = 0; k < HID; ++k) {
    float v = h1[(long)i * HID + k];
    z0 += v * W2[k * NCLS + 0];
    z1 += v * W2[k * NCLS + 1];
  }
  g[(long)i * NCLS + 0] = z0;
  g[(long)i * NCLS + 1] = z1;
}

__global__ void agg2_kernel(const int* __restrict__ row, const int* __restrict__ col,
                            const float* __restrict__ w, const float* __restrict__ dinv,
                            const float* __restrict__ g, float* __restrict__ agg, int E) {
  int e = blockIdx.x * blockDim.x + threadIdx.x;
  if (e >= E) return;
  int s = row[e], t = col[e];
  float coef = dinv[s] * w[e] * dinv[t];
  atomicAdd(&agg[(long)t * NCLS + 0], coef * g[(long)s * NCLS + 0]);
  atomicAdd(&agg[(long)t * NCLS + 1], coef * g[(long)s * NCLS + 1]);
}

// out = log_softmax(agg2 + dinv^2*g + b2) over the 2 classes
__global__ void final_kernel(const float* __restrict__ agg2, const float* __restrict__ dinv,
                             const float* __restrict__ g, const float* __restrict__ b2,
                             float* __restrict__ out, int n) {
  int i = blockIdx.x * blockDim.x + threadIdx.x;
  if (i >= n) return;
  float di2 = dinv[i] * dinv[i];
  float z0 = agg2[(long)i * NCLS + 0] + di2 * g[(long)i * NCLS + 0] + b2[0];
  float z1 = agg2[(long)i * NCLS + 1] + di2 * g[(long)i * NCLS + 1] + b2[1];
  float m = fmaxf(z0, z1);
  float lse = m + logf(expf(z0 - m) + expf(z1 - m));
  out[(long)i * NCLS + 0] = z0 - lse;
  out[(long)i * NCLS + 1] = z1 - lse;
}

extern "C" void kernel_launch(void* const* d_in, const int* in_sizes, int n_in,
                              void* d_out, int out_size, void* d_ws, size_t ws_size,
                              hipStream_t stream) {
  const float* x  = (const float*)d_in[0];
  const int*   ei = (const int*)d_in[1];
  const float* ew = (const float*)d_in[2];
  const float* W1 = (const float*)d_in[3];
  const float* b1 = (const float*)d_in[4];
  const float* W2 = (const float*)d_in[5];
  const float* b2 = (const float*)d_in[6];
  float* out = (float*)d_out;

  const int N = in_sizes[0] / FIN;
  const int E = in_sizes[2];
  const int* row = ei;       // edge_index[0] = source (gather)
  const int* col = ei + E;   // edge_index[1] = target (scatter)

  // workspace layout (floats); atomic targets first so one zero pass covers them
  float* ws   = (float*)d_ws;
  float* dinv = ws;                    // [N]   deg -> D^-1/2 in place
  float* agg1 = ws + N;                // [16N] layer-1 scatter accumulator
  float* agg2 = ws + (long)17 * N;     // [2N]  layer-2 scatter accumulator
  float* h    = ws + (long)19 * N;     // [16N] x@W1, then relu'd h1 in place
  float* g    = ws + (long)35 * N;     // [2N]  h1@W2
  // total: 37N floats (~14.8 MB for N=100k)

  const int B = 256;
  long zn = (long)19 * N;
  zero_kernel<<<(int)((zn + B - 1) / B), B, 0, stream>>>(ws, zn);
  deg_kernel<<<(E + B - 1) / B, B, 0, stream>>>(col, ew, dinv, E);
  dinv_kernel<<<(N + B - 1) / B, B, 0, stream>>>(dinv, N);

  int nTiles = N / 16;
  if (nTiles > 0)
    xw1_wmma_kernel<<<(nTiles * 32 + B - 1) / B, B, 0, stream>>>(x, W1, h, nTiles);
  int rem = N - nTiles * 16;
  if (rem > 0)
    xw1_tail_kernel<<<(rem * 16 + B - 1) / B, B, 0, stream>>>(x, W1, h, nTiles * 16, N);

  agg1_kernel<<<(E + B - 1) / B, B, 0, stream>>>(row, col, ew, dinv, h, agg1, E);
  long n16 = (long)N * HID;
  relu_bias_kernel<<<(int)((n16 + B - 1) / B), B, 0, stream>>>(agg1, dinv, b1, h, n16);
  gw2_kernel<<<(N + B - 1) / B, B, 0, stream>>>(h, W2, g, N);
  agg2_kernel<<<(E + B - 1) / B, B, 0, stream>>>(row, col, ew, dinv, g, agg2, E);
  final_kernel<<<(N + B - 1) / B, B, 0, stream>>>(agg2, dinv, g, b2, out, N);
}